// NoiseDiscriminator_46480136077525
// MI455X (gfx1250) — compile-verified
//
#include <hip/hip_runtime.h>

#define DIM 64
#define CHUNK 256

typedef float v2f __attribute__((ext_vector_type(2)));
typedef float v8f __attribute__((ext_vector_type(8)));

// global / LDS address-space pointer types for the async builtin
typedef __attribute__((address_space(1))) int gint_t;
typedef __attribute__((address_space(3))) int lint_t;

#if defined(__has_builtin)
#if __has_builtin(__builtin_amdgcn_global_load_async_to_lds_b32) && __has_builtin(__builtin_amdgcn_s_wait_asynccnt)
#define HAVE_ASYNC_LDS 1
#else
#define HAVE_ASYNC_LDS 0
#endif
#if __has_builtin(__builtin_amdgcn_wmma_f32_16x16x4_f32)
#define HAVE_WMMA_F32 1
#else
#define HAVE_WMMA_F32 0
#endif
#else
#define HAVE_ASYNC_LDS 0
#define HAVE_WMMA_F32 0
#endif

__device__ __forceinline__ void gadd(float* p, float v) {
  // hardware no-return global_atomic_add_f32 path
  unsafeAtomicAdd(p, v);
}

// ---------------- SpMM: y[dst] += val * x[src] ----------------
// Edge metadata staged into LDS (async-to-LDS on CDNA5), then one
// 16-lane group per edge moves a 256B row with a single b128 load
// and 4 f32 atomics per lane.
__global__ __launch_bounds__(256) void spmm_kernel(
    const float* __restrict__ x, float* __restrict__ y,
    const float* __restrict__ vals, const int* __restrict__ src,
    const int* __restrict__ dst, int nEdges) {
  __shared__ int   s_src[CHUNK];
  __shared__ int   s_dst[CHUNK];
  __shared__ float s_val[CHUNK];

  const int t    = threadIdx.x;
  const int base = blockIdx.x * CHUNK;
  const int e    = base + t;

#if HAVE_ASYNC_LDS
  if (e < nEdges) {
    __builtin_amdgcn_global_load_async_to_lds_b32(
        (gint_t*)(int*)(src + e), (lint_t*)(&s_src[t]), 0, 0);
    __builtin_amdgcn_global_load_async_to_lds_b32(
        (gint_t*)(int*)(dst + e), (lint_t*)(&s_dst[t]), 0, 0);
    __builtin_amdgcn_global_load_async_to_lds_b32(
        (gint_t*)(float*)(vals + e), (lint_t*)(&s_val[t]), 0, 0);
  }
  __builtin_amdgcn_s_wait_asynccnt(0);
  __syncthreads();
#else
  if (e < nEdges) {
    s_src[t] = src[e];
    s_dst[t] = dst[e];
    s_val[t] = vals[e];
  }
  __syncthreads();
#endif

  const int g = t >> 4;   // 16 groups of 16 lanes; group g handles edges g, g+16, ...
  const int l = t & 15;   // lane within group: 4 floats of the 64-float row

  #pragma unroll 4
  for (int i = 0; i < 16; ++i) {
    const int ei = i * 16 + g;
    if (base + ei < nEdges) {
      const int   s = s_src[ei];
      const int   d = s_dst[ei];
      const float v = s_val[ei];
      const float4 xv = ((const float4*)(x + (size_t)s * DIM))[l];
      float* yp = y + (size_t)d * DIM + (size_t)l * 4;
      gadd(yp + 0, v * xv.x);
      gadd(yp + 1, v * xv.y);
      gadd(yp + 2, v * xv.z);
      gadd(yp + 3, v * xv.w);
    }
  }
}

// ---------------- concat [emb_user; emb_item] -> x ----------------
__global__ __launch_bounds__(256) void concat_kernel(
    const float4* __restrict__ eu, const float4* __restrict__ ei,
    float4* __restrict__ x, int nU4, int nT4) {
  int i = blockIdx.x * 256 + threadIdx.x;
  if (i < nT4) x[i] = (i < nU4) ? eu[i] : ei[i - nU4];
}

__global__ __launch_bounds__(256) void zero_kernel(float4* __restrict__ y, int nT4) {
  int i = blockIdx.x * 256 + threadIdx.x;
  if (i < nT4) y[i] = make_float4(0.f, 0.f, 0.f, 0.f);
}

// ---------------- per-batch accumulator: acc += layer[gathered rows] ----------------
__global__ __launch_bounds__(256) void gather_kernel(
    const float* __restrict__ xsrc, const int* __restrict__ users,
    const int* __restrict__ items, float* __restrict__ accU,
    float* __restrict__ accI, int B, int nUsers, int accumulate) {
  int idx = blockIdx.x * 256 + threadIdx.x;
  int half = B * DIM;
  if (idx < half) {
    int b = idx >> 6, d = idx & 63;
    float v = xsrc[(size_t)users[b] * DIM + d];
    accU[idx] = accumulate ? (accU[idx] + v) : v;
  } else if (idx < 2 * half) {
    int j = idx - half;
    int b = j >> 6, d = j & 63;
    float v = xsrc[(size_t)(nUsers + items[b]) * DIM + d];
    accI[j] = accumulate ? (accI[j] + v) : v;
  }
}

// ---------------- gamma[b] = dot(accU[b], accI[b]) * scale ----------------
// One wave per 16 batch rows: diag of (16x64)·(64x16) via v_wmma_f32_16x16x4_f32,
// K accumulated in 16 steps of 4. Diagonal extraction per ISA C/D layout:
// lane r holds (r,r) in c[r]; lane r+24 holds (r+8,r+8) in c[r].
__global__ __launch_bounds__(32) void dot_kernel(
    const float* __restrict__ accU, const float* __restrict__ accI,
    float* __restrict__ out, int B, float scale) {
  const int lane = threadIdx.x;
  const int b0 = blockIdx.x * 16;
#if HAVE_WMMA_F32
  const int M = lane & 15;
  int row = b0 + M;
  if (row >= B) row = B - 1;            // keep EXEC all-ones (WMMA requirement)
  const int koff = (lane < 16) ? 0 : 2; // A layout: VGPR0 = K0/K2, VGPR1 = K1/K3
  const float* up = accU + (size_t)row * DIM;
  const float* vp = accI + (size_t)row * DIM;
  v8f c = {};
  #pragma unroll
  for (int k = 0; k < DIM; k += 4) {
    v2f a, b;
    a[0] = up[k + koff];
    a[1] = up[k + koff + 1];
    b[0] = vp[k + koff];
    b[1] = vp[k + koff + 1];
    c = __builtin_amdgcn_wmma_f32_16x16x4_f32(
        /*neg_a=*/false, a, /*neg_b=*/false, b,
        /*c_mod=*/(short)0, c, /*reuse_a=*/false, /*reuse_b=*/false);
  }
  if (lane < 8) {
    if (b0 + lane < B) out[b0 + lane] = c[lane] * scale;
  } else if (lane >= 24) {
    int r = lane - 24;
    if (b0 + 8 + r < B) out[b0 + 8 + r] = c[r] * scale;
  }
#else
  if (lane < 16 && b0 + lane < B) {
    const float* up = accU + (size_t)(b0 + lane) * DIM;
    const float* vp = accI + (size_t)(b0 + lane) * DIM;
    float s = 0.f;
    for (int k = 0; k < DIM; ++k) s += up[k] * vp[k];
    out[b0 + lane] = s * scale;
  }
#endif
}

extern "C" void kernel_launch(void* const* d_in, const int* in_sizes, int n_in,
                              void* d_out, int out_size, void* d_ws, size_t ws_size,
                              hipStream_t stream) {
  const float* emb_user = (const float*)d_in[0];
  const float* emb_item = (const float*)d_in[1];
  const float* vals     = (const float*)d_in[2];
  const int*   src      = (const int*)d_in[3];
  const int*   dst      = (const int*)d_in[4];
  const int*   users    = (const int*)d_in[5];
  const int*   items    = (const int*)d_in[6];
  float*       out      = (float*)d_out;

  const int nUsersRows = in_sizes[0] / DIM;   // 50000
  const int nItemsRows = in_sizes[1] / DIM;   // 100000
  const int nNodes     = nUsersRows + nItemsRows;
  const int nEdges     = in_sizes[2];         // 2.4M
  const int B          = in_sizes[5];         // 4096

  // ws layout: xbuf | ybuf | accU | accI  (~79 MB total)
  size_t nodeFloats = (size_t)nNodes * DIM;
  float* xb   = (float*)d_ws;
  float* yb   = xb + nodeFloats;
  float* accU = yb + nodeFloats;
  float* accI = accU + (size_t)B * DIM;

  const int nT4 = (int)(nodeFloats / 4);
  const int nU4 = nUsersRows * DIM / 4;
  dim3 blk(256);
  const int copyGrid   = (nT4 + 255) / 256;
  const int gatherGrid = (2 * B * DIM + 255) / 256;
  const int spmmGrid   = (nEdges + CHUNK - 1) / CHUNK;

  // layer 0 embedding (concat) + accumulator init
  concat_kernel<<<copyGrid, blk, 0, stream>>>(
      (const float4*)emb_user, (const float4*)emb_item, (float4*)xb, nU4, nT4);
  gather_kernel<<<gatherGrid, blk, 0, stream>>>(
      xb, users, items, accU, accI, B, nUsersRows, 0);

  // 3 propagation layers, ping-pong buffers
  float* cur = xb;
  float* nxt = yb;
  for (int layer = 0; layer < 3; ++layer) {
    zero_kernel<<<copyGrid, blk, 0, stream>>>((float4*)nxt, nT4);
    spmm_kernel<<<spmmGrid, blk, 0, stream>>>(cur, nxt, vals, src, dst, nEdges);
    gather_kernel<<<gatherGrid, blk, 0, stream>>>(
        nxt, users, items, accU, accI, B, nUsersRows, 1);
    float* tmp = cur; cur = nxt; nxt = tmp;
  }

  // gamma = dot(accU, accI) / (L+1)^2
  const int dotGrid = (B + 15) / 16;
  dot_kernel<<<dotGrid, dim3(32), 0, stream>>>(accU, accI, out, B, 1.0f / 16.0f);
}